// spatial_attention_66855460930231
// MI455X (gfx1250) — compile-verified
//
#include <hip/hip_runtime.h>
#include <hip/hip_bf16.h>
#include <cstdint>

// Problem constants (from reference): x [16, 64, 256, 256] f32, w [1,2,3,3] f32.
#define BATCH   16
#define CH      64
#define HH      256
#define WW      256
#define TW      32            // tile width  (interior)
#define THT     16            // tile height (interior)
#define RW      (TW + 2)      // 34  region width incl. halo
#define RH      (THT + 2)     // 18  region height incl. halo
#define RSZ     (RW * RH)     // 612 region positions
#define NTHREADS 512
#define NBUF    8             // async staging ring (prefetch distance 4)
#define PLANE   (HH * WW)         // 65536 elements per channel plane
#define PLANE_B (PLANE * 4)       // 262144 bytes
#define BSTRIDE ((size_t)CH * PLANE)

typedef float v4f __attribute__((ext_vector_type(4)));   // native clang vector

// CDNA5 async global -> LDS copy (ASYNCcnt path). vdst = LDS byte offset in a
// VGPR, vaddr = 64-bit global address pair, saddr = off.
__device__ __forceinline__ void async_load_f32(uint32_t lds_off, uint64_t gaddr) {
  asm volatile("global_load_async_to_lds_b32 %0, %1, off"
               :: "v"(lds_off), "v"(gaddr) : "memory");
}
#define WAITA(n) asm volatile("s_wait_asynccnt " #n ::: "memory")

__global__ __launch_bounds__(NTHREADS)
void spatial_attention_66855460930231_kernel(const float* __restrict__ x,
                                             const float* __restrict__ w,
                                             float* __restrict__ out) {
  __shared__ __align__(16) float xbuf[NBUF][RSZ]; // async staging ring (19.6 KB)
  __shared__ float fmax_s[RSZ];                   // channel-max feature
  __shared__ float favg_s[RSZ];                   // channel-avg feature

  const int tid   = threadIdx.x;
  const int tx    = tid & (TW - 1);   // 0..31
  const int ty    = tid >> 5;         // 0..15
  const int tileX = blockIdx.x;
  const int tileY = blockIdx.y;
  const int b     = blockIdx.z;

  const float* xb = x + (size_t)b * BSTRIDE;

  // Region position ownership: p0 = tid (all), p1 = tid + 512 (tid < 100).
  const int  p0   = tid;
  const int  p1   = tid + NTHREADS;
  const bool has1 = (p1 < RSZ);
  const int  p1e  = has1 ? p1 : p0;

  const int rx0 = p0 % RW, ry0 = p0 / RW;
  const int rh0 = tileY * THT - 1 + ry0;
  const int rw0 = tileX * TW  - 1 + rx0;
  const bool inb0 = ((unsigned)rh0 < HH) && ((unsigned)rw0 < WW);
  const int rh0c = min(max(rh0, 0), HH - 1);
  const int rw0c = min(max(rw0, 0), WW - 1);

  const int rx1 = p1e % RW, ry1 = p1e / RW;
  const int rh1 = tileY * THT - 1 + ry1;
  const int rw1 = tileX * TW  - 1 + rx1;
  const bool inb1 = ((unsigned)rh1 < HH) && ((unsigned)rw1 < WW);
  const int rh1c = min(max(rh1, 0), HH - 1);
  const int rw1c = min(max(rw1, 0), WW - 1);

  const uint64_t ga0 = (uint64_t)(uintptr_t)(xb + rh0c * WW + rw0c);
  const uint64_t ga1 = (uint64_t)(uintptr_t)(xb + rh1c * WW + rw1c);
  const uint32_t la0 = (uint32_t)(uintptr_t)&xbuf[0][p0];
  const uint32_t la1 = (uint32_t)(uintptr_t)&xbuf[0][p1e];
  const uint32_t bufStride = (uint32_t)(RSZ * sizeof(float));

  float s0 = 0.f, m0 = -INFINITY, s1 = 0.f, m1 = -INFINITY;

  // --- Phase 1: channel reduction, async-pipelined through LDS -------------
  // Wave-uniform split: waves 0-3 own two region slots (2 issues/channel),
  // waves 4-15 own one. Per-wave wait counts must match per-wave issue counts.
  if (tid < 128) {
    auto issue2 = [&](int c) {
      const uint32_t lb = (uint32_t)(c & (NBUF - 1)) * bufStride;
      const uint64_t go = (uint64_t)(uint32_t)c * PLANE_B;
      async_load_f32(la0 + lb, ga0 + go);
      async_load_f32(la1 + lb, ga1 + go);
    };
    auto consume2 = [&](int c) {
      const float v0 = xbuf[c & (NBUF - 1)][p0];
      const float v1 = xbuf[c & (NBUF - 1)][p1e];
      s0 += v0; m0 = fmaxf(m0, v0);
      s1 += v1; m1 = fmaxf(m1, v1);
    };
    issue2(0); issue2(1); issue2(2); issue2(3);
#pragma unroll 2
    for (int c = 0; c < CH - 4; ++c) {   // c = 0..59
      issue2(c + 4);
      WAITA(8);                          // 4 newer channels x 2 still in flight
      consume2(c);
    }
    WAITA(6); consume2(60);
    WAITA(4); consume2(61);
    WAITA(2); consume2(62);
    WAITA(0); consume2(63);
  } else {
    auto issue1 = [&](int c) {
      const uint32_t lb = (uint32_t)(c & (NBUF - 1)) * bufStride;
      async_load_f32(la0 + lb, ga0 + (uint64_t)(uint32_t)c * PLANE_B);
    };
    auto consume1 = [&](int c) {
      const float v0 = xbuf[c & (NBUF - 1)][p0];
      s0 += v0; m0 = fmaxf(m0, v0);
    };
    issue1(0); issue1(1); issue1(2); issue1(3);
#pragma unroll 2
    for (int c = 0; c < CH - 4; ++c) {
      issue1(c + 4);
      WAITA(4);
      consume1(c);
    }
    WAITA(3); consume1(60);
    WAITA(2); consume1(61);
    WAITA(1); consume1(62);
    WAITA(0); consume1(63);
  }

  // feat: channel0 = max, channel1 = avg (matches reference concat order);
  // 0 outside image = conv zero padding.
  fmax_s[p0] = inb0 ? m0 : 0.f;
  favg_s[p0] = inb0 ? s0 * (1.f / CH) : 0.f;
  if (has1) {
    fmax_s[p1] = inb1 ? m1 : 0.f;
    favg_s[p1] = inb1 ? s1 * (1.f / CH) : 0.f;
  }
  __syncthreads();

  // --- Phase 2: 3x3 conv + sigmoid -----------------------------------------
  float wk[18];
#pragma unroll
  for (int i = 0; i < 18; ++i) wk[i] = w[i];   // uniform -> scalar loads

  float a = 0.f;
#pragma unroll
  for (int kh = 0; kh < 3; ++kh) {
#pragma unroll
    for (int kw = 0; kw < 3; ++kw) {
      const int p = (ty + kh) * RW + (tx + kw);
      a = fmaf(wk[kh * 3 + kw],     fmax_s[p], a);
      a = fmaf(wk[9 + kh * 3 + kw], favg_s[p], a);
    }
  }
  const float attn = 1.f / (1.f + __expf(-a));

  // Publish attn for the tile (overlay on xbuf: async traffic fully retired).
  float* attn_s = &xbuf[0][0];               // 512 floats, 16B-aligned
  attn_s[ty * TW + tx] = attn;
  __syncthreads();

  // --- Phase 3: broadcast multiply, b128 vectorized ------------------------
  // Remap: q = pixel quad (8 quads/row x 16 rows), cg = 16-channel group.
  // x loads stay RT (L2 reuse from phase 1); output stores are non-temporal
  // so the 268 MB write-once stream doesn't evict the x tiles from L2.
  const int q  = tid & 127;
  const int cg = tid >> 7;                   // 0..3
  const int qy = q >> 3;
  const int qx = q & 7;
  const int h  = tileY * THT + qy;
  const int wq = tileX * TW + qx * 4;

  const v4f av = *(const v4f*)&attn_s[qy * TW + qx * 4];
  const float* xp = xb + (size_t)(cg * 16) * PLANE + h * WW + wq;
  float*       op = out + (size_t)b * BSTRIDE + (size_t)(cg * 16) * PLANE + h * WW + wq;

#pragma unroll
  for (int k = 0; k < 16; ++k) {             // offsets k*256KB fit imm24
    const v4f xv = *(const v4f*)(xp + (size_t)k * PLANE);
    const v4f ov = av * xv;                  // component-wise
    __builtin_nontemporal_store(ov, (v4f*)(op + (size_t)k * PLANE));
  }
}

extern "C" void kernel_launch(void* const* d_in, const int* in_sizes, int n_in,
                              void* d_out, int out_size, void* d_ws, size_t ws_size,
                              hipStream_t stream) {
  (void)in_sizes; (void)n_in; (void)out_size; (void)d_ws; (void)ws_size;
  const float* x = (const float*)d_in[0];
  const float* w = (const float*)d_in[1];
  float*     out = (float*)d_out;
  dim3 grid(WW / TW, HH / THT, BATCH);   // (8, 16, 16)
  spatial_attention_66855460930231_kernel<<<grid, NTHREADS, 0, stream>>>(x, w, out);
}